// BaseMovingLayer_43688407335336
// MI455X (gfx1250) — compile-verified
//
#include <hip/hip_runtime.h>

// MEGA MultiHeadEMA as a chunked linear scan with WMMA f32 GEMMs.
// B=4, D=1024, L=4096, N=16.  Chunk T=16 -> 256 chunks per channel.
//
//   h_n[l] = q_n h_n[l-1] + u_n x[l],  out[l] = sum_n g_n h_n[l] + omega x[l]
//   q = 1 - sig(delta)*sig(alpha), u = sig(delta)*sig(alpha)*beta, g = gamma/4
//
// Chunked:  Z[c,n]   = sum_j q^{15-j} x[cT+j]                 (GEMM, WMMA)
//           H[c]     = q^16 H[c-1] + u Z[c]                   (serial scan)
//           out[c,i] = (X Tt)[c,i] + (Hprev M1)[c,i] + w x    (GEMMs, WMMA)
//   Tt[j][i] = kloc[i-j] (i>=j), kloc[m] = sum_n g_n u_n q^m
//   M1[n][i] = g_n q^{i+1}

typedef float v2f __attribute__((ext_vector_type(2)));
typedef float v8f __attribute__((ext_vector_type(8)));

#define BDIM   1024
#define LSEQ   4096
#define NCH    16
#define CHUNKS 256
#define LSTR   17   // padded LDS row stride (17 coprime with 64 banks)

__device__ __forceinline__ float sigmoidf_(float v) {
    return 1.0f / (1.0f + __expf(-v));
}

__global__ __launch_bounds__(256)
void ema_scan_wmma(const float* __restrict__ x,
                   const float* __restrict__ delta,
                   const float* __restrict__ alpha,
                   const float* __restrict__ beta,
                   const float* __restrict__ gamma,
                   const float* __restrict__ omega,
                   float* __restrict__ out)
{
    __shared__ float sX[CHUNKS * LSTR];   // x reshaped [chunk][i]
    __shared__ float sZ[CHUNKS * LSTR];   // Z, then Hprev after scan
    __shared__ float sV [NCH * LSTR];     // V[n][j]  = q_n^{15-j}
    __shared__ float sM1[NCH * LSTR];     // M1[n][i] = g_n q_n^{i+1}
    __shared__ float sTt[NCH * LSTR];     // Tt[j][i] = kloc[i-j] (i>=j)
    __shared__ float sW [NCH];            // g_n * u_n
    __shared__ float sQT[NCH];            // q_n^16
    __shared__ float sU [NCH];            // u_n

    const int bid = blockIdx.x;           // b*D + d
    const int d   = bid & (BDIM - 1);
    const int tid = threadIdx.x;
    const long base = (long)bid * LSEQ;

    // ---- load x[b,d,:] into LDS (coalesced), build coefficient tables ----
    for (int idx = tid; idx < LSEQ; idx += 256) {
        float v = x[base + idx];
        sX[(idx >> 4) * LSTR + (idx & 15)] = v;
    }
    {
        const int n = tid >> 4;
        const int j = tid & 15;
        float de = delta[d * NCH + n];
        float al = alpha[d * NCH + n];
        float be = beta [d * NCH + n];
        float ga = gamma[d * NCH + n];
        float p = sigmoidf_(de) * sigmoidf_(al);
        float q = 1.0f - p;
        float u = p * be;
        float g = ga * 0.25f;             // scale = sqrt(1/16)
        float pw = 1.0f;                  // q^(15-j)
        for (int e = 0; e < 15 - j; ++e) pw *= q;
        sV[n * LSTR + j] = pw;
        float pw2 = q;                    // q^(j+1)
        for (int e = 0; e < j; ++e) pw2 *= q;
        sM1[n * LSTR + j] = g * pw2;
        if (j == 0) {
            float q16 = 1.0f;
            for (int e = 0; e < 16; ++e) q16 *= q;
            sQT[n] = q16;
            sU [n] = u;
            sW [n] = g * u;
        }
    }
    __syncthreads();

    // ---- transposed lower-triangular Toeplitz: Tt[j][i] = kloc[i-j] ----
    {
        const int j = tid >> 4;
        const int i = tid & 15;
        float s = 0.0f;
        if (i >= j) {
            const int m = i - j;          // q_n^m == sV[n][15-m]
            for (int n = 0; n < NCH; ++n)
                s += sW[n] * sV[n * LSTR + (15 - m)];
        }
        sTt[j * LSTR + i] = s;
    }
    __syncthreads();

    const int wave = tid >> 5;
    const int lane = tid & 31;
    const int ml = lane & 15;             // A-row / B-col / D-col index
    const int kh = lane >> 4;             // K half-select

    // ---- phase 2: Z[256x16] = Xc[256x16] * V^T[16x16] (K=16 = 4 WMMA) ----
    for (int t = 0; t < 2; ++t) {
        const int c0 = (wave * 2 + t) * 16;
        v8f acc = {};
        for (int kk = 0; kk < 4; ++kk) {
            const int k0 = kk * 4 + 2 * kh;
            v2f a, b;
            a.x = sX[(c0 + ml) * LSTR + k0];
            a.y = sX[(c0 + ml) * LSTR + k0 + 1];
            b.x = sV[ml * LSTR + k0];      // B[k][n] = V[n][k]
            b.y = sV[ml * LSTR + k0 + 1];
            acc = __builtin_amdgcn_wmma_f32_16x16x4_f32(
                      false, a, false, b, (short)0, acc, false, false);
        }
        for (int r = 0; r < 8; ++r)
            sZ[(c0 + r + 8 * kh) * LSTR + ml] = acc[r];
    }
    __syncthreads();

    // ---- phase 3: serial inter-chunk scan (wave 0, lanes 0..15) ----
    // Hprev[c] = H[c-1];  H[c] = q^16 H[c-1] + u Z[c]
    if (tid < 16) {
        const float qT = sQT[tid];
        const float u  = sU [tid];
        float H = 0.0f;
        for (int c = 0; c < CHUNKS; ++c) {
            float zc = sZ[c * LSTR + tid];
            sZ[c * LSTR + tid] = H;        // becomes Hprev[c][n]
            H = qT * H + u * zc;
        }
    }
    __syncthreads();

    // ---- phase 4: out = Xc*Tt + Hprev*M1 + omega*x ----
    const float om = omega[d];
    for (int t = 0; t < 2; ++t) {
        const int c0 = (wave * 2 + t) * 16;
        v8f acc = {};
        for (int kk = 0; kk < 4; ++kk) {   // intra-chunk Toeplitz term
            const int k0 = kk * 4 + 2 * kh;
            v2f a, b;
            a.x = sX[(c0 + ml) * LSTR + k0];
            a.y = sX[(c0 + ml) * LSTR + k0 + 1];
            b.x = sTt[k0 * LSTR + ml];
            b.y = sTt[(k0 + 1) * LSTR + ml];
            acc = __builtin_amdgcn_wmma_f32_16x16x4_f32(
                      false, a, false, b, (short)0, acc, false, false);
        }
        for (int kk = 0; kk < 4; ++kk) {   // carry-in term
            const int k0 = kk * 4 + 2 * kh;
            v2f a, b;
            a.x = sZ[(c0 + ml) * LSTR + k0];      // A[m][k] = Hprev[c0+m][n=k]
            a.y = sZ[(c0 + ml) * LSTR + k0 + 1];
            b.x = sM1[k0 * LSTR + ml];            // B[n][i] = M1[n][i]
            b.y = sM1[(k0 + 1) * LSTR + ml];
            acc = __builtin_amdgcn_wmma_f32_16x16x4_f32(
                      false, a, false, b, (short)0, acc, false, false);
        }
        for (int r = 0; r < 8; ++r) {
            const int c = c0 + r + 8 * kh;
            float v = acc[r] + om * sX[c * LSTR + ml];
            out[base + c * 16 + ml] = v;   // coalesced: lanes 0..15 contiguous
        }
    }
}

extern "C" void kernel_launch(void* const* d_in, const int* in_sizes, int n_in,
                              void* d_out, int out_size, void* d_ws, size_t ws_size,
                              hipStream_t stream) {
    const float* x     = (const float*)d_in[0];
    const float* delta = (const float*)d_in[1];
    const float* alpha = (const float*)d_in[2];
    const float* beta  = (const float*)d_in[3];
    const float* gamma = (const float*)d_in[4];
    const float* omega = (const float*)d_in[5];
    float* out = (float*)d_out;

    const int n_channels = in_sizes[0] / LSEQ;   // B*D = 4096
    hipLaunchKernelGGL(ema_scan_wmma, dim3(n_channels), dim3(256), 0, stream,
                       x, delta, alpha, beta, gamma, omega, out);
}